// DescriptorModule_6373731467699
// MI455X (gfx1250) — compile-verified
//
#include <hip/hip_runtime.h>

typedef __attribute__((ext_vector_type(2))) float v2f;
typedef __attribute__((ext_vector_type(8))) float v8f;

#define SS 8
#define NN 2048
#define TT 64
#define DHID 32
#define DEMB 64
#define SUBF 16
#define SEG0 24           // first 24 neighbors -> net type, rest -> net type+1
#define WPB  8            // waves per block (wave32), one atom per wave

// D = A(16x4) * B(4x16) + C, full fp32 (matches reference precision).
__device__ __forceinline__ v8f wmma_f32(v2f a, v2f b, v8f c) {
  return __builtin_amdgcn_wmma_f32_16x16x4_f32(
      /*neg_a=*/false, a, /*neg_b=*/false, b,
      /*c_mod=*/(short)0, c, /*reuse_a=*/false, /*reuse_b=*/false);
}

__global__ __launch_bounds__(256)
void descriptor_kernel(const float* __restrict__ pos,       // S*N*3
                       const float* __restrict__ W1g,       // 3*1*32
                       const float* __restrict__ B1g,       // 3*32
                       const float* __restrict__ W2g,       // 3*32*64
                       const float* __restrict__ B2g,       // 3*64
                       const int*   __restrict__ types,     // 1*N
                       const long long* __restrict__ neigh, // S*N*T (int64)
                       float* __restrict__ out)             // S*N*64*16
{
  __shared__ float W1s[3 * DHID];
  __shared__ float B1s[3 * DHID];
  __shared__ float W2T[3 * DEMB * DHID];   // [m][e][c]  (transposed: c contiguous)
  __shared__ float B2s[3 * DEMB];
  __shared__ float SV [WPB][TT];           // s[t]
  __shared__ float RTT[WPB][4 * TT];       // [a][t]  (t contiguous)
  __shared__ float QLT[WPB][2][4 * DHID];  // [seg][a][c]  (c contiguous)
  __shared__ float UL [WPB][2][4];         // u_seg[a] = sum_t r_tilde[t][a]
  __shared__ float PLT[WPB][4 * DEMB];     // [a][e]  (e contiguous)

  const int tid = threadIdx.x;

  // ---- cooperative weight load into LDS (shared by all 8 atoms) ----
  for (int i = tid; i < 3 * DHID * DEMB; i += 256) {
    const int m = i >> 11;            // net
    const int r = i & 2047;
    const int c = r >> 6;             // row   (DHID)
    const int e = r & 63;             // col   (DEMB)
    W2T[(m << 11) + (e << 5) + c] = W2g[i];
  }
  if (tid < 3 * DHID) { W1s[tid] = W1g[tid]; B1s[tid] = B1g[tid]; }
  if (tid < 3 * DEMB) { B2s[tid] = B2g[tid]; }
  __syncthreads();

  const int w    = tid >> 5;
  const int lane = tid & 31;
  const int atom = blockIdx.x * WPB + w;
  const int si   = atom / NN;
  const int ni   = atom % NN;

  const int type = types[ni];
  const int mA = type;        // PAIR_IDX[type][0]
  const int mB = type + 1;    // PAIR_IDX[type][1]

  const float xi = pos[(si * NN + ni) * 3 + 0];
  const float yi = pos[(si * NN + ni) * 3 + 1];
  const float zi = pos[(si * NN + ni) * 3 + 2];

  // ---- phase 1: neighbor geometry (2 neighbors per lane) ----
  const long long* nl = neigh + (long long)(si * NN + ni) * TT;
  #pragma unroll
  for (int r = 0; r < 2; ++r) {
    const int t = lane + 32 * r;
    const int j = (int)nl[t];
    float dx = pos[(si * NN + j) * 3 + 0] - xi;
    float dy = pos[(si * NN + j) * 3 + 1] - yi;
    float dz = pos[(si * NN + j) * 3 + 2] - zi;
    dx -= rintf(dx); dy -= rintf(dy); dz -= rintf(dz);   // LENGTH == 1, RNE
    float dist = sqrtf(dx * dx + dy * dy + dz * dz);
    dist = fmaxf(dist, 1e-12f);
    float sv = 1.0f / dist;
    const float uu = (dist - 0.3f) * (1.0f / 0.3f);
    const float fac = (dist < 0.3f) ? 1.0f
                    : ((dist < 0.6f) ? (0.5f * cosf(3.14159265358979323846f * uu) + 0.5f)
                                     : 0.0f);
    sv *= fac;
    const float inv = sv / dist;
    SV [w][t]          = sv;
    RTT[w][0 * TT + t] = sv;
    RTT[w][1 * TT + t] = dx * inv;
    RTT[w][2 * TT + t] = dy * inv;
    RTT[w][3 * TT + t] = dz * inv;
  }
  __syncthreads();

  const int nlo = lane & 15;       // tile N / tile M row index
  const int hi  = lane >> 4;       // lane half -> K pair {0,1} vs {2,3}
  const int klo = hi * 2;
  const int a4c = nlo & 3;         // clamped 'a' index: lanes 4..15 fetch
                                   // duplicate columns (harmless: WMMA columns
                                   // are independent; only a<4 is consumed)

  // ---- phase 2: q_seg = h_seg^T(32xT) @ r_tilde(Tx4) via WMMA ----
  // seg0: t in [0,24) -> kt 0..5 ; seg1: t in [24,64) -> kt 6..15 (K=4-aligned)
  #pragma unroll
  for (int seg = 0; seg < 2; ++seg) {
    const int m   = (seg == 0) ? mA : mB;
    const int kt0 = (seg == 0) ? 0 : 6;
    const int kt1 = (seg == 0) ? 6 : 16;
    #pragma unroll
    for (int mc = 0; mc < 2; ++mc) {
      const int   c   = 16 * mc + nlo;
      const float w1c = W1s[m * DHID + c];
      const float b1c = B1s[m * DHID + c];
      v8f acc = {};
      #pragma unroll
      for (int kt = kt0; kt < kt1; ++kt) {
        const int t0 = 4 * kt + klo;
        v2f a, b;
        a.x = fmaxf(SV[w][t0 + 0] * w1c + b1c, 0.0f);  // h(t,c) in A layout
        a.y = fmaxf(SV[w][t0 + 1] * w1c + b1c, 0.0f);
        b.x = RTT[w][a4c * TT + t0 + 0];               // contiguous pair -> b64
        b.y = RTT[w][a4c * TT + t0 + 1];
        acc = wmma_f32(a, b, acc);
      }
      if (nlo < 4) {                                    // one exec toggle, 8
        #pragma unroll                                  // consecutive stores
        for (int jj = 0; jj < 8; ++jj) {
          QLT[w][seg][nlo * DHID + 16 * mc + jj + 8 * hi] = acc[jj];
        }
      }
    }
  }

  // ---- u_seg[a] = sum_t r_tilde[t][a] (for the b2 rank-1 bias term) ----
  {
    const int a4 = lane & 3;
    const int ck = lane >> 2;             // 8 chunks of 8 t-values
    float p0 = 0.0f, p1 = 0.0f;
    #pragma unroll
    for (int q = 0; q < 8; ++q) {
      const int t = 8 * ck + q;
      const float v = RTT[w][a4 * TT + t];
      if (t < SEG0) p0 += v; else p1 += v;
    }
    p0 += __shfl_xor(p0, 4);  p1 += __shfl_xor(p1, 4);
    p0 += __shfl_xor(p0, 8);  p1 += __shfl_xor(p1, 8);
    p0 += __shfl_xor(p0, 16); p1 += __shfl_xor(p1, 16);
    if (lane < 4) { UL[w][0][lane] = p0; UL[w][1][lane] = p1; }
  }
  __syncthreads();

  // ---- phase 3: P(64x4) = sum_seg W2[m]^T(64x32) @ q_seg(32x4) + b2 x u ----
  v8f cp[4] = {{}, {}, {}, {}};
  #pragma unroll
  for (int seg = 0; seg < 2; ++seg) {
    const int m = (seg == 0) ? mA : mB;
    #pragma unroll
    for (int kc = 0; kc < 8; ++kc) {
      const int c0 = 4 * kc + klo;
      v2f b;                             // B[k,a] = q[c0+k][a], contiguous pair
      b.x = QLT[w][seg][a4c * DHID + c0 + 0];
      b.y = QLT[w][seg][a4c * DHID + c0 + 1];
      #pragma unroll
      for (int me = 0; me < 4; ++me) {
        v2f a;                           // A[e,c] = W2[m][c][e], contiguous pair
        a.x = W2T[(m << 11) + (16 * me + nlo) * DHID + c0 + 0];
        a.y = W2T[(m << 11) + (16 * me + nlo) * DHID + c0 + 1];
        cp[me] = wmma_f32(a, b, cp[me]);
      }
    }
  }
  {
    const float uA = UL[w][0][a4c];      // duplicate cols for lanes 4..15: unused
    const float uB = UL[w][1][a4c];
    #pragma unroll
    for (int me = 0; me < 4; ++me) {
      #pragma unroll
      for (int jj = 0; jj < 8; ++jj) {
        const int e = 16 * me + jj + 8 * hi;
        cp[me][jj] += B2s[mA * DEMB + e] * uA + B2s[mB * DEMB + e] * uB;
      }
    }
  }
  if (nlo < 4) {
    #pragma unroll
    for (int me = 0; me < 4; ++me) {
      #pragma unroll
      for (int jj = 0; jj < 8; ++jj) {
        PLT[w][nlo * DEMB + 16 * me + jj + 8 * hi] = cp[me][jj];
      }
    }
  }
  __syncthreads();

  // ---- phase 4: D(64x16) = P(64x4) @ (P[:16,:])^T(4x16) via WMMA ----
  float* outp = out + (long long)atom * (DEMB * SUBF);
  v2f bR;                                // B[k,f] = P[f][k]
  bR.x = PLT[w][(klo + 0) * DEMB + nlo];
  bR.y = PLT[w][(klo + 1) * DEMB + nlo];
  #pragma unroll
  for (int md = 0; md < 4; ++md) {
    v2f a;                               // A[e,k] = P[16*md+e][k]
    a.x = PLT[w][(klo + 0) * DEMB + 16 * md + nlo];
    a.y = PLT[w][(klo + 1) * DEMB + 16 * md + nlo];
    v8f zero = {};
    v8f d = wmma_f32(a, bR, zero);
    #pragma unroll
    for (int jj = 0; jj < 8; ++jj) {
      const int e = 16 * md + jj + 8 * hi;
      outp[e * SUBF + nlo] = d[jj];      // D[s,n,e,f], f = nlo; 64B-contiguous
    }
  }
}

extern "C" void kernel_launch(void* const* d_in, const int* in_sizes, int n_in,
                              void* d_out, int out_size, void* d_ws, size_t ws_size,
                              hipStream_t stream) {
  (void)in_sizes; (void)n_in; (void)out_size; (void)d_ws; (void)ws_size;
  const float*     pos = (const float*)d_in[0];
  const float*     W1  = (const float*)d_in[1];
  const float*     b1  = (const float*)d_in[2];
  const float*     W2  = (const float*)d_in[3];
  const float*     b2  = (const float*)d_in[4];
  const int*       ty  = (const int*)d_in[5];
  const long long* nlg = (const long long*)d_in[6];
  float* out = (float*)d_out;

  dim3 grid((SS * NN) / WPB);   // 2048 blocks x 256 threads (8 wave32 = 8 atoms)
  descriptor_kernel<<<grid, 256, 0, stream>>>(pos, W1, b1, W2, b2, ty, nlg, out);
}